// PTV3MaskFormer3D_69054484185753
// MI455X (gfx1250) — compile-verified
//
#include <hip/hip_runtime.h>
#include <hip/hip_bf16.h>

#define B_   2
#define N_   131072
#define K_   20
#define E_   256
#define TEMP_INV 20.0f          // 1 / 0.05
#define SCORE_THR 0.04f
#define FSTR 260                // LDS row stride (floats): conflict-free (bank = 4r+k), 16B aligned
#define SSTR 260
#define FROWB (FSTR * 4)        // 1040 bytes per LDS feature row

typedef float v2f __attribute__((ext_vector_type(2)));
typedef float v8f __attribute__((ext_vector_type(8)));

// ---------------------------------------------------------------------------
// Kernel 1: slot path.  x = w1@slots^T + b1 ; GroupNorm(4 groups) ; ReLU ;
// slot2 = w2@x + b2 ; sn = l2norm(slot2 over E).  One block per batch.
// Output snw layout: (B, K, E) row-major.
// ---------------------------------------------------------------------------
__global__ __launch_bounds__(256) void slot_path_kernel(
    const float* __restrict__ slots, const float* __restrict__ w1,
    const float* __restrict__ b1,    const float* __restrict__ gn_g,
    const float* __restrict__ gn_b,  const float* __restrict__ w2,
    const float* __restrict__ b2,    float* __restrict__ snw)
{
  __shared__ float sl[K_ * E_];   // slots (k-major); later reused for slot2 (f-major)
  __shared__ float xr[E_ * K_];   // relu(gn(x)) : [e][k]
  __shared__ float red[256], red2[256];
  __shared__ float nrm[K_];

  const int b = blockIdx.x;
  const int t = threadIdx.x;

  for (int i = t; i < K_ * E_; i += 256) sl[i] = slots[(size_t)b * K_ * E_ + i];
  __syncthreads();

  // x[o=t][k] = b1[o] + sum_e slots[k][e] * w1[o][e]
  float acc[K_];
  {
    const float bb = b1[t];
#pragma unroll
    for (int k = 0; k < K_; k++) acc[k] = bb;
    const float* wrow = w1 + (size_t)t * E_;
    for (int e = 0; e < E_; e++) {
      const float w = wrow[e];
#pragma unroll
      for (int k = 0; k < K_; k++) acc[k] = fmaf(sl[k * E_ + e], w, acc[k]);
    }
  }
  // GroupNorm stats over (64 channels x 20 slots) per group (group = t>>6)
  float s = 0.f, s2 = 0.f;
#pragma unroll
  for (int k = 0; k < K_; k++) { s += acc[k]; s2 += acc[k] * acc[k]; }
  red[t] = s; red2[t] = s2;
  __syncthreads();
  for (int off = 32; off >= 1; off >>= 1) {
    if ((t & 63) < off) { red[t] += red[t + off]; red2[t] += red2[t + off]; }
    __syncthreads();
  }
  const float mu  = red[(t >> 6) << 6] * (1.0f / 1280.0f);
  const float var = red2[(t >> 6) << 6] * (1.0f / 1280.0f) - mu * mu;
  const float rsv = rsqrtf(var + 1e-5f);
  const float g = gn_g[t], be = gn_b[t];
#pragma unroll
  for (int k = 0; k < K_; k++) {
    float v = (acc[k] - mu) * rsv * g + be;
    xr[t * K_ + k] = v > 0.f ? v : 0.f;
  }
  __syncthreads();

  // slot2[f=t][k] = b2[f] + sum_e xr[e][k] * w2[f][e]
  float a2[K_];
  {
    const float bb = b2[t];
#pragma unroll
    for (int k = 0; k < K_; k++) a2[k] = bb;
    const float* wrow = w2 + (size_t)t * E_;
    for (int e = 0; e < E_; e++) {
      const float w = wrow[e];
#pragma unroll
      for (int k = 0; k < K_; k++) a2[k] = fmaf(xr[e * K_ + k], w, a2[k]);
    }
  }
#pragma unroll
  for (int k = 0; k < K_; k++) sl[t * K_ + k] = a2[k];  // reuse sl as slot2[f][k]
  __syncthreads();

  if (t < K_) {
    float ss = 0.f;
    for (int f = 0; f < E_; f++) { const float v = sl[f * K_ + t]; ss = fmaf(v, v, ss); }
    nrm[t] = fmaxf(sqrtf(ss), 1e-12f);
  }
  __syncthreads();

  for (int k = 0; k < K_; k++)
    snw[((size_t)b * K_ + k) * E_ + t] = sl[t * K_ + k] / nrm[k];
}

// ---------------------------------------------------------------------------
// Async DMA of one 16-row (16 x 1KB) feature tile into an LDS buffer with
// padded rows (1040B).  32 x GLOBAL_LOAD_ASYNC_TO_LDS_B128, tracked on
// ASYNCcnt.  Per instruction: 32 lanes x 16B = 512B (half a row); the
// offset:512 immediate advances BOTH the global and LDS addresses, which is
// consistent within a row.
// ---------------------------------------------------------------------------
__device__ __forceinline__ void prefetch_tile(const float* fbase, int pbase,
                                              unsigned int ldsrow0, int lane16)
{
#pragma unroll
  for (int r = 0; r < 16; ++r) {
    const unsigned int lds = ldsrow0 + (unsigned int)(r * FROWB);
    const int goff = (pbase + r) * (E_ * 4) + lane16;
    asm volatile(
        "global_load_async_to_lds_b128 %0, %1, %2\n\t"
        "global_load_async_to_lds_b128 %0, %1, %2 offset:512"
        :: "v"(lds), "v"(goff), "s"(fbase) : "memory");
  }
}

// ---------------------------------------------------------------------------
// Kernel 2: fused normalize + logits GEMM (WMMA f32 16x16x4) + softmax +
// per-slot score accumulation.  One wave per block; 16-point tiles,
// double-buffered via async-to-LDS (no barriers needed).  Mask is written
// directly into d_out in the final (B, K, N) layout.
// ---------------------------------------------------------------------------
__global__ __launch_bounds__(32) void logits_softmax_kernel(
    const float* __restrict__ features, const float* __restrict__ snw,
    unsigned long long* __restrict__ scoreQ, float* __restrict__ outMask)
{
  __shared__ float snl[K_ * SSTR];        // 20 x 260
  __shared__ float ftile[2 * 16 * FSTR];  // double-buffered 16 x 260

  const int b    = blockIdx.y;
  const int lane = threadIdx.x;    // 0..31, single wave
  const int p    = lane & 15;      // point (B col) and slot (A row) index
  const int h    = lane >> 4;      // k-pair select

  // stage sn (20 x 256) into LDS, float4-coalesced
  for (int i = lane * 4; i < K_ * E_; i += 32 * 4) {
    const float4 v = *reinterpret_cast<const float4*>(snw + (size_t)b * K_ * E_ + i);
    const int r = i >> 8, c = i & 255;
    float* d = &snl[r * SSTR + c];
    d[0] = v.x; d[1] = v.y; d[2] = v.z; d[3] = v.w;
  }
  __syncthreads();   // single-wave WG: S_NOP, kept as a compiler fence

  float scAcc[12];
#pragma unroll
  for (int j = 0; j < 12; j++) scAcc[j] = 0.f;

  const int   k0   = 2 * h;
  const int   m2r  = (p < 4) ? (16 + p) : 0;     // A rows 16..19 valid, rest zeroed
  const float mskv = (p < 4) ? 1.0f : 0.0f;
  const float* a0row = &snl[p * SSTR];
  const float* a1row = &snl[m2r * SSTR];

  const float* fbase = features + (size_t)b * N_ * E_;
  const unsigned int ldsft = (unsigned int)(size_t)(void*)ftile;  // low 32b = LDS offset
  const int lane16 = lane * 16;

  const int ntiles = N_ / 16;          // 8192
  const int stride = gridDim.x;        // 2048

  int cur = 0;
  prefetch_tile(fbase, blockIdx.x * 16, ldsft, lane16);

  for (int tile = blockIdx.x; tile < ntiles; tile += stride) {
    const int nxt = tile + stride;
    if (nxt < ntiles) {
      prefetch_tile(fbase, nxt * 16, ldsft + (unsigned int)((cur ^ 1) * 16 * FROWB), lane16);
      asm volatile("s_wait_asynccnt 0x20" ::: "memory");  // current tile's 32 done
    } else {
      asm volatile("s_wait_asynccnt 0x0" ::: "memory");
    }

    const float* frow = &ftile[(cur * 16 + p) * FSTR];
    v8f c0 = {0.f, 0.f, 0.f, 0.f, 0.f, 0.f, 0.f, 0.f};
    v8f c1 = {0.f, 0.f, 0.f, 0.f, 0.f, 0.f, 0.f, 0.f};
    float ssq = 0.f;

#pragma unroll 4
    for (int kk = 0; kk < E_; kk += 4) {
      v2f bf = *reinterpret_cast<const v2f*>(&frow[kk + k0]);   // B[k][point]
      ssq = fmaf(bf.x, bf.x, ssq);
      ssq = fmaf(bf.y, bf.y, ssq);
      v2f a0 = *reinterpret_cast<const v2f*>(&a0row[kk + k0]);  // A rows 0..15
      v2f a1 = *reinterpret_cast<const v2f*>(&a1row[kk + k0]);  // A rows 16..31
      a1 = a1 * mskv;
      c0 = __builtin_amdgcn_wmma_f32_16x16x4_f32(false, a0, false, bf,
                                                 (short)0, c0, false, false);
      c1 = __builtin_amdgcn_wmma_f32_16x16x4_f32(false, a1, false, bf,
                                                 (short)0, c1, false, false);
    }

    // lanes p and p+16 together cover all 256 dims of point p
    ssq += __shfl_xor(ssq, 16, 32);
    // 20/max(sqrt(ssq),1e-12) == 20*rsqrt(max(ssq,1e-24)); v_rsq_f32 is ~1ulp
    const float scale = TEMP_INV * __builtin_amdgcn_rsqf(fmaxf(ssq, 1e-24f));

    // this lane holds, for point p: slots (i+8h) in c0[i]; slots 16+i (h==0,i<4)
    // in c1[i].  Invalid lanes get -3e38 so every softmax loop has a constant
    // bound (no register-indexed lg[] -> no m0/v_movrels loops).
    float lg[12];
#pragma unroll
    for (int i = 0; i < 8; i++) lg[i] = c0[i] * scale;
#pragma unroll
    for (int i = 0; i < 4; i++) lg[8 + i] = (h == 0) ? c1[i] * scale : -3.0e38f;

    float mx = lg[0];
#pragma unroll
    for (int j = 1; j < 12; j++) mx = fmaxf(mx, lg[j]);
    mx = fmaxf(mx, __shfl_xor(mx, 16, 32));
    float sum = 0.f;
#pragma unroll
    for (int j = 0; j < 12; j++) { lg[j] = __expf(lg[j] - mx); sum += lg[j]; }
    sum += __shfl_xor(sum, 16, 32);
    const float inv = __builtin_amdgcn_rcpf(sum);

    const int gp = tile * 16 + p;
#pragma unroll
    for (int j = 0; j < 8; j++) {
      const int slot = j + 8 * h;
      const float mv = lg[j] * inv;
      outMask[((size_t)b * K_ + slot) * N_ + gp] = mv;
      scAcc[j] += mv;
    }
    if (h == 0) {
#pragma unroll
      for (int j = 0; j < 4; j++) {
        const float mv = lg[8 + j] * inv;
        outMask[((size_t)b * K_ + (16 + j)) * N_ + gp] = mv;
        scAcc[8 + j] += mv;
      }
    }
    cur ^= 1;
  }

  // reduce score partials across the 16 lanes of each half-wave
#pragma unroll
  for (int j = 0; j < 12; j++)
    for (int off = 1; off < 16; off <<= 1)
      scAcc[j] += __shfl_xor(scAcc[j], off, 32);

  if (p == 0) {   // lane 0 -> slots 0..7,16..19 ; lane 16 -> slots 8..15
    for (int j = 0; j < 8; j++)
      atomicAdd(&scoreQ[b * 32 + (j + 8 * h)],
                (unsigned long long)llrintf(scAcc[j] * 1048576.0f));
    if (h == 0)
      for (int j = 0; j < 4; j++)
        atomicAdd(&scoreQ[b * 32 + 16 + j],
                  (unsigned long long)llrintf(scAcc[8 + j] * 1048576.0f));
  }
}

// ---------------------------------------------------------------------------
// Kernel 3: tiny NMS.  Softmax rows sum to 1 => at most one entry > 0.5 per
// point => inter is diagonal => iou off-diagonal == 0 => IOU suppression is a
// no-op; kept == (score >= thr) in stable-descending-score order.
// Emits per-slot destination column dst[s] (or -1) and kept count.
// ---------------------------------------------------------------------------
__global__ void nms_kernel(const unsigned long long* __restrict__ scoreQ,
                           int* __restrict__ dstb)
{
  const int b = blockIdx.x;
  if (threadIdx.x != 0) return;
  float scores[K_];
  for (int sI = 0; sI < K_; sI++)
    scores[sI] = (float)((double)scoreQ[b * 32 + sI] * (1.0 / 1048576.0) / (double)N_);
  int order[K_];
  for (int i = 0; i < K_; i++) order[i] = i;
  for (int i = 1; i < K_; i++) {           // stable insertion sort, descending
    const int oi = order[i];
    const float sc = scores[oi];
    int j = i - 1;
    while (j >= 0 && scores[order[j]] < sc) { order[j + 1] = order[j]; j--; }
    order[j + 1] = oi;
  }
  int dst[K_];
  for (int i = 0; i < K_; i++) dst[i] = -1;
  int c = 0;
  for (int j = 0; j < K_; j++) {
    const int sI = order[j];
    if (scores[sI] >= SCORE_THR) dst[sI] = c++;
  }
  for (int sI = 0; sI < K_; sI++) dstb[b * 32 + sI] = dst[sI];
  dstb[b * 32 + 24] = c;   // number of kept columns
}

// ---------------------------------------------------------------------------
// Kernel 4: in-place column permute + row renormalize of d_out (B,K,N).
// Each thread owns one (b,n) column: read all 20, then scatter-write with
// uniform runtime column indices (no register-indexed arrays -> no scratch).
// ---------------------------------------------------------------------------
__global__ __launch_bounds__(256) void remap_kernel(float* __restrict__ out,
                                                    const int* __restrict__ dstb)
{
  const int b = blockIdx.y;
  const int n = blockIdx.x * 256 + threadIdx.x;
  float m[K_];
#pragma unroll
  for (int sI = 0; sI < K_; sI++) m[sI] = out[((size_t)b * K_ + sI) * N_ + n];
  int ds[K_];
#pragma unroll
  for (int sI = 0; sI < K_; sI++) ds[sI] = dstb[b * 32 + sI];
  const int nk = dstb[b * 32 + 24];
  float rs = 0.f;
#pragma unroll
  for (int sI = 0; sI < K_; sI++) if (ds[sI] >= 0) rs += m[sI];
  const float inv = 1.0f / (rs + 1e-8f);
#pragma unroll
  for (int sI = 0; sI < K_; sI++)
    if (ds[sI] >= 0) out[((size_t)b * K_ + ds[sI]) * N_ + n] = m[sI] * inv;
  for (int c = nk; c < K_; c++) out[((size_t)b * K_ + c) * N_ + n] = 0.f;
}

// ---------------------------------------------------------------------------
extern "C" void kernel_launch(void* const* d_in, const int* in_sizes, int n_in,
                              void* d_out, int out_size, void* d_ws, size_t ws_size,
                              hipStream_t stream) {
  (void)in_sizes; (void)n_in; (void)out_size; (void)ws_size;
  const float* features = (const float*)d_in[0];
  const float* slots    = (const float*)d_in[1];
  const float* w1       = (const float*)d_in[2];
  const float* b1       = (const float*)d_in[3];
  const float* gn_g     = (const float*)d_in[4];
  const float* gn_b     = (const float*)d_in[5];
  const float* w2       = (const float*)d_in[6];
  const float* b2       = (const float*)d_in[7];
  float* out = (float*)d_out;

  float*              snw    = (float*)d_ws;                               // 40960 B
  unsigned long long* scoreQ = (unsigned long long*)((char*)d_ws + 40960); // 512 B
  int*                dstb   = (int*)((char*)d_ws + 41472);                // 256 B

  hipMemsetAsync((char*)d_ws + 40960, 0, 512, stream);
  slot_path_kernel<<<dim3(B_), 256, 0, stream>>>(slots, w1, b1, gn_g, gn_b, w2, b2, snw);
  logits_softmax_kernel<<<dim3(2048, B_), 32, 0, stream>>>(features, snw, scoreQ, out);
  nms_kernel<<<dim3(B_), 32, 0, stream>>>(scoreQ, dstb);
  remap_kernel<<<dim3(N_ / 256, B_), 256, 0, stream>>>(out, dstb);
}